// CSDE_decoder_54709293416510
// MI455X (gfx1250) — compile-verified
//
#include <hip/hip_runtime.h>
#include <hip/hip_bf16.h>
#include <math.h>

// Problem constants (match reference)
#define BSZ 1024
#define NN  325
#define HH_ 64
#define TT  12
#define MROWS (BSZ * NN)          // 332800 = 16 * 20800

typedef __attribute__((ext_vector_type(2))) float v2f;
typedef __attribute__((ext_vector_type(8))) float v8f;

// ---------------------------------------------------------------------------
// Diffusion MLP: gval[b,n] = tanh(relu(relu(relu(y@GW0+b0)@GW1+b1)@GW2+b2)@GWf+bf)
// One block per batch row b; 64 threads; everything staged in LDS.
// ---------------------------------------------------------------------------
__global__ __launch_bounds__(64)
void gfunc_kernel(const float* __restrict__ y,
                  const float* __restrict__ GW0, const float* __restrict__ Gb0,
                  const float* __restrict__ GW1, const float* __restrict__ Gb1,
                  const float* __restrict__ GW2, const float* __restrict__ Gb2,
                  const float* __restrict__ GWf, const float* __restrict__ Gbf,
                  float* __restrict__ gval)
{
    __shared__ float sy[NN];
    __shared__ float sh[HH_];
    __shared__ float sh2[HH_];

    const int b   = blockIdx.x;
    const int tid = threadIdx.x;

    for (int n = tid; n < NN; n += 64) sy[n] = y[(size_t)b * NN + n];
    __syncthreads();

    // layer 0: [NN] -> [64], thread = output channel
    float acc = Gb0[tid];
    for (int n = 0; n < NN; ++n) acc = fmaf(sy[n], GW0[n * HH_ + tid], acc);
    sh[tid] = fmaxf(acc, 0.f);
    __syncthreads();

    // layer 1
    acc = Gb1[tid];
    #pragma unroll 8
    for (int k = 0; k < HH_; ++k) acc = fmaf(sh[k], GW1[k * HH_ + tid], acc);
    sh2[tid] = fmaxf(acc, 0.f);
    __syncthreads();

    // layer 2 (write back into sh)
    acc = Gb2[tid];
    #pragma unroll 8
    for (int k = 0; k < HH_; ++k) acc = fmaf(sh2[k], GW2[k * HH_ + tid], acc);
    sh[tid] = fmaxf(acc, 0.f);
    __syncthreads();

    // final: [64] -> [NN], tanh
    for (int n = tid; n < NN; n += 64) {
        float s = Gbf[n];
        #pragma unroll 8
        for (int k = 0; k < HH_; ++k) s = fmaf(sh[k], GWf[k * NN + n], s);
        gval[(size_t)b * NN + n] = tanhf(s);
    }
}

// ---------------------------------------------------------------------------
// Drift + SDE update for one step t.
//   a = z@Wa[:64] + y*Wa[64] + ba ; b likewise ; glu = a*sigmoid(b)
//   f = glu@wo + bo ; y_next = y + f*dt + gval*sqrt(dt)*eps
// 8 waves / block, each wave owns a 16-row tile of M = B*N rows.
// K=64 handled as 16 x V_WMMA_F32_16X16X4_F32 steps per column tile (4 tiles).
// Weights staged in LDS pre-swizzled into B-fragment order:
//   sWP[j*128 + col*2 + r] = W[2j + r][col]   (j = K-pair index, r in {0,1})
// so each B fragment is a single aligned ds_load_b64 at an immediate offset.
// ---------------------------------------------------------------------------
__global__ __launch_bounds__(256)
void drift_step_kernel(const float* __restrict__ z,     // [M,64]
                       const float* __restrict__ Wa,    // [65,64] (this step)
                       const float* __restrict__ ba,    // [64]
                       const float* __restrict__ Wb,    // [65,64]
                       const float* __restrict__ bb,    // [64]
                       const float* __restrict__ wo,    // [64]
                       const float* __restrict__ bo,    // [1]
                       const float* __restrict__ times, // [T+1]
                       const float* __restrict__ yin,   // [M]
                       const float* __restrict__ gval,  // [M]
                       const float* __restrict__ eps,   // [M] (this step)
                       float* __restrict__ yout)        // [M]
{
    __shared__ float sWaP[32 * 128];   // paired-row fragment layout, 16 KB
    __shared__ float sWbP[32 * 128];
    __shared__ float sWaL[64];         // row 64 (rank-1 y term)
    __shared__ float sWbL[64];
    __shared__ float swo[64];
    __shared__ float sba[64];
    __shared__ float sbb[64];

    const int tid = threadIdx.x;
    for (int i = tid; i < 64 * 64; i += 256) {
        const int k   = i >> 6;
        const int col = i & 63;
        const int dst = (k >> 1) * 128 + col * 2 + (k & 1);
        sWaP[dst] = Wa[i];
        sWbP[dst] = Wb[i];
    }
    if (tid < 64) {
        sWaL[tid] = Wa[64 * 64 + tid];
        sWbL[tid] = Wb[64 * 64 + tid];
        swo[tid]  = wo[tid];
        sba[tid]  = ba[tid];
        sbb[tid]  = bb[tid];
    }
    __syncthreads();

    const int wave = tid >> 5;          // 0..7
    const int lane = tid & 31;
    const int lo   = lane & 15;
    const int hi   = lane >> 4;         // 0 or 1

    const int tileRow = (blockIdx.x * 8 + wave) * 16;   // exact cover: 2600*8*16 = M

    const float* zrow = z + (size_t)(tileRow + lo) * 64 + hi * 2;
    const float* pA   = &sWaP[hi * 128 + lo * 2];
    const float* pB   = &sWbP[hi * 128 + lo * 2];

    v8f accA[4] = {v8f{}, v8f{}, v8f{}, v8f{}};
    v8f accB[4] = {v8f{}, v8f{}, v8f{}, v8f{}};

    #pragma unroll 4
    for (int kk = 0; kk < 16; ++kk) {
        const v2f afrag = *(const v2f*)(zrow + kk * 4);
        #pragma unroll
        for (int ct = 0; ct < 4; ++ct) {
            const v2f bfa = *(const v2f*)(pA + kk * 256 + ct * 32);
            const v2f bfb = *(const v2f*)(pB + kk * 256 + ct * 32);
            accA[ct] = __builtin_amdgcn_wmma_f32_16x16x4_f32(
                false, afrag, false, bfa, (short)0, accA[ct], false, false);
            accB[ct] = __builtin_amdgcn_wmma_f32_16x16x4_f32(
                false, afrag, false, bfb, (short)0, accB[ct], false, false);
        }
    }

    // epilogue: rank-1 y correction + bias + GLU + wo reduction + SDE update
    const float dt   = times[1] - times[0];
    const float sqdt = sqrtf(dt);
    const float bo0  = bo[0];
    const float yv   = yin[tileRow + lo];   // y for row (tileRow+lo); halves duplicate

    #pragma unroll
    for (int v = 0; v < 8; ++v) {
        const int mloc = v + 8 * hi;                 // row within tile for this lane's C elems
        const float ym = __shfl(yv, mloc, 32);       // y[tileRow + mloc]
        float p = 0.f;
        #pragma unroll
        for (int ct = 0; ct < 4; ++ct) {
            const int col = ct * 16 + lo;
            const float av  = accA[ct][v] + sba[col] + ym * sWaL[col];
            const float bv  = accB[ct][v] + sbb[col] + ym * sWbL[col];
            const float glu = av * (1.f / (1.f + __expf(-bv)));
            p = fmaf(glu, swo[col], p);
        }
        // reduce over the 16 'lo' lanes (xor on bits 0..3 stays within the half)
        p += __shfl_xor(p, 1, 32);
        p += __shfl_xor(p, 2, 32);
        p += __shfl_xor(p, 4, 32);
        p += __shfl_xor(p, 8, 32);
        if (lo == 0) {
            const int m  = tileRow + mloc;
            const float f = p + bo0;
            yout[m] = ym + f * dt + gval[m] * sqdt * eps[m];
        }
    }
}

// ---------------------------------------------------------------------------
// Launch: 12 sequential steps; y ping-pongs through workspace, final into d_out
// ---------------------------------------------------------------------------
extern "C" void kernel_launch(void* const* d_in, const int* in_sizes, int n_in,
                              void* d_out, int out_size, void* d_ws, size_t ws_size,
                              hipStream_t stream)
{
    const float* times = (const float*)d_in[0];
    const float* y0    = (const float*)d_in[1];
    const float* z     = (const float*)d_in[2];
    const float* noise = (const float*)d_in[3];   // [T,B,N]
    const float* Wa    = (const float*)d_in[4];   // [T,65,64]
    const float* ba    = (const float*)d_in[5];   // [T,64]
    const float* Wb    = (const float*)d_in[6];
    const float* bb    = (const float*)d_in[7];
    const float* Wo    = (const float*)d_in[8];   // [T,64]
    const float* bo    = (const float*)d_in[9];   // [T]
    const float* GW0   = (const float*)d_in[10];
    const float* Gb0   = (const float*)d_in[11];
    const float* GW1   = (const float*)d_in[12];
    const float* Gb1   = (const float*)d_in[13];
    const float* GW2   = (const float*)d_in[14];
    const float* Gb2   = (const float*)d_in[15];
    const float* GWf   = (const float*)d_in[16];
    const float* Gbf   = (const float*)d_in[17];

    float* yA = (float*)d_ws;
    float* yB = yA + MROWS;
    float* gv = yB + MROWS;

    const float* ycur = y0;
    for (int t = 0; t < TT; ++t) {
        gfunc_kernel<<<BSZ, 64, 0, stream>>>(ycur, GW0, Gb0, GW1, Gb1, GW2, Gb2,
                                             GWf, Gbf, gv);
        float* ynext = (t == TT - 1) ? (float*)d_out : ((t & 1) ? yB : yA);
        drift_step_kernel<<<MROWS / (8 * 16), 256, 0, stream>>>(
            z,
            Wa + (size_t)t * 65 * 64, ba + (size_t)t * 64,
            Wb + (size_t)t * 65 * 64, bb + (size_t)t * 64,
            Wo + (size_t)t * 64, bo + t,
            times, ycur, gv, noise + (size_t)t * MROWS, ynext);
        ycur = ynext;
    }
}